// GCNNet_74251394613508
// MI455X (gfx1250) — compile-verified
//
#include <hip/hip_runtime.h>
#include <hip/hip_bf16.h>

// ---------------------------------------------------------------------------
// GCN (2-layer) for MI455X / gfx1250.
//   h1   = x @ W1                     (WMMA f32 16x16x4, K=128, 2 N-tiles,
//                                      B fragments pre-packed + LDS-staged)
//   a1   = scatter(norm * h1[src])    (b128 gather + global_atomic_add_f32)
//   h2   = relu(a1 + dinv^2*h1 + b1)  (self-loop folded, no atomics)
//   h3   = h2 @ W2                    (WMMA f32 16x16x4, K=32, 1 N-tile)
//   out  = scatter(norm * h3[src]) + dinv^2*h3 + b2
// ---------------------------------------------------------------------------

typedef __attribute__((ext_vector_type(2))) float v2f;
typedef __attribute__((ext_vector_type(8))) float v8f;

#define IN_C 128
#define HID_C 32
#define OUT_C 16

// ---------------- degree / normalization ----------------------------------

__global__ __launch_bounds__(256) void fill_ones(float* __restrict__ p, int n) {
  int i = blockIdx.x * blockDim.x + threadIdx.x;
  if (i < n) p[i] = 1.0f;                       // self-loop contributes 1 to deg
}

__global__ __launch_bounds__(256) void deg_accum(const long long* __restrict__ dst,
                                                 float* __restrict__ deg,
                                                 long long nE) {
  long long i = (long long)blockIdx.x * blockDim.x + threadIdx.x;
  if (i < nE) atomicAdd(&deg[dst[i]], 1.0f);
}

__global__ __launch_bounds__(256) void rsqrt_inplace(float* __restrict__ p, int n) {
  int i = blockIdx.x * blockDim.x + threadIdx.x;
  if (i < n) p[i] = __frsqrt_rn(p[i]);          // deg >= 1 always (self-loops)
}

// ---------------- pack W into WMMA B-fragment order ------------------------
// Wp layout: [K/4][NTILES][lane 0..31][2 floats]  ->  lane's v2f is one b64.
// B 4x16 fragment (mirror of ISA 7.12.2 A layout): lanes 0-15 hold K=0,1 of
// column N=lane; lanes 16-31 hold K=2,3 of column N=lane-16.

template <int K, int NTILES>
__global__ __launch_bounds__(256) void pack_wmma_b(const float* __restrict__ W,
                                                   float* __restrict__ Wp) {
  constexpr int NC = NTILES * 16;
  constexpr int KSTEPS = K / 4;
  const int idx = blockIdx.x * blockDim.x + threadIdx.x;   // over KSTEPS*NTILES*32
  if (idx >= KSTEPS * NTILES * 32) return;
  const int lane = idx & 31;
  const int t    = (idx >> 5) % NTILES;
  const int ks   = idx / (32 * NTILES);
  const int m    = lane & 15;
  const int koff = (lane >> 4) * 2;
  const int col  = t * 16 + m;
  Wp[idx * 2 + 0] = W[(ks * 4 + koff) * NC + col];
  Wp[idx * 2 + 1] = W[(ks * 4 + koff + 1) * NC + col];
}

// ---------------- fp32 WMMA GEMM: H[n_rows x NT*16] = X[n_rows x K] @ W ----
// A 16x4 (ISA 7.12.2): lanes 0-15 hold K=0,1; lanes 16-31 hold K=2,3 -> b64.
// B fed from LDS (packed fragments) -> one ds_load_b64 per (k-step, tile).
// C/D 16x16: VGPR r -> row r (lanes 0-15) / row r+8 (lanes 16-31).

template <int K, int NTILES>
__global__ __launch_bounds__(256) void gcn_gemm_wmma(const float* __restrict__ X,
                                                     const float* __restrict__ Wp,
                                                     float* __restrict__ H,
                                                     int n_rows) {
  constexpr int KSTEPS = K / 4;
  constexpr int NC = NTILES * 16;
  constexpr int WELEMS = KSTEPS * NTILES * 64;   // == K * NC floats
  __shared__ float lwp[WELEMS];

  // cooperative LDS stage of packed W (all threads participate, then barrier)
  for (int i = threadIdx.x * 4; i < WELEMS; i += blockDim.x * 4) {
    *(float4*)&lwp[i] = *(const float4*)&Wp[i];
  }
  __syncthreads();

  const int wave = threadIdx.x >> 5;
  const int lane = threadIdx.x & 31;
  const int rowBase = (blockIdx.x * (blockDim.x >> 5) + wave) * 16;
  if (rowBase >= n_rows) return;                // wave-uniform: EXEC stays full

  const int m    = lane & 15;
  const int hi   = lane >> 4;                   // half-wave select
  const int koff = hi * 2;                      // this lane's K sub-offset

  int rowA = rowBase + m;
  if (rowA >= n_rows) rowA = n_rows - 1;        // clamp loads, keep EXEC full
  const float* xrow = X + (long long)rowA * K;

  v8f acc[NTILES] = {};
  for (int ks = 0; ks < KSTEPS; ++ks) {
    v2f a;
    a.x = xrow[ks * 4 + koff];                  // pair -> global_load_b64
    a.y = xrow[ks * 4 + koff + 1];
#pragma unroll
    for (int t = 0; t < NTILES; t++) {
      v2f b = *(const v2f*)&lwp[(ks * NTILES + t) * 64 + lane * 2];  // ds b64
      acc[t] = __builtin_amdgcn_wmma_f32_16x16x4_f32(
          /*neg_a=*/false, a, /*neg_b=*/false, b,
          /*c_mod=*/(short)0, acc[t], /*reuse_a=*/false, /*reuse_b=*/false);
    }
  }

  if (rowBase + 16 <= n_rows) {                 // full tile: unguarded stores
#pragma unroll
    for (int t = 0; t < NTILES; t++)
#pragma unroll
      for (int r = 0; r < 8; r++)
        H[(long long)(rowBase + r + hi * 8) * NC + t * 16 + m] = acc[t][r];
  } else {                                      // tail tile (never hit: N%16==0)
#pragma unroll
    for (int t = 0; t < NTILES; t++)
#pragma unroll
      for (int r = 0; r < 8; r++) {
        const int row = rowBase + r + hi * 8;
        if (row < n_rows) H[(long long)row * NC + t * 16 + m] = acc[t][r];
      }
  }
}

// ---------------- edge gather / scaled scatter-add -------------------------
// One thread per (edge, 4-channel group): b128 gather, 4x global_atomic_add_f32.

template <int CG>  // channel groups of 4; C = CG*4, CG power of two
__global__ __launch_bounds__(256) void edge_aggregate(const float* __restrict__ h,
                                                      const long long* __restrict__ src,
                                                      const long long* __restrict__ dst,
                                                      const float* __restrict__ dinv,
                                                      float* __restrict__ agg,
                                                      long long nE) {
  const long long idx = (long long)blockIdx.x * blockDim.x + threadIdx.x;
  if (idx >= nE * CG) return;
  const long long e = idx / CG;                 // power-of-two -> shift
  const int g = (int)(idx & (CG - 1));
  const long long s = src[e];
  const long long d = dst[e];
  const float nrm = dinv[s] * dinv[d];
  const float4 v = *reinterpret_cast<const float4*>(h + s * (CG * 4) + g * 4);
  float* out = agg + d * (CG * 4) + g * 4;
  atomicAdd(out + 0, v.x * nrm);
  atomicAdd(out + 1, v.y * nrm);
  atomicAdd(out + 2, v.z * nrm);
  atomicAdd(out + 3, v.w * nrm);
}

// ---------------- epilogues (self-loop + bias [+ relu]) --------------------

__global__ __launch_bounds__(256) void epilogue_l1(float* __restrict__ agg,
                                                   const float* __restrict__ h1,
                                                   const float* __restrict__ dinv,
                                                   const float* __restrict__ b1,
                                                   int n_nodes) {
  const int i = blockIdx.x * blockDim.x + threadIdx.x;
  if (i >= n_nodes * HID_C) return;
  const int node = i >> 5;                      // HID_C == 32
  const int c = i & (HID_C - 1);
  const float d = dinv[node];
  const float v = agg[i] + h1[i] * d * d + b1[c];
  agg[i] = v > 0.0f ? v : 0.0f;
}

__global__ __launch_bounds__(256) void epilogue_l2(float* __restrict__ out,
                                                   const float* __restrict__ h3,
                                                   const float* __restrict__ dinv,
                                                   const float* __restrict__ b2,
                                                   int n_nodes) {
  const int i = blockIdx.x * blockDim.x + threadIdx.x;
  if (i >= n_nodes * OUT_C) return;
  const int node = i >> 4;                      // OUT_C == 16
  const int c = i & (OUT_C - 1);
  const float d = dinv[node];
  out[i] = out[i] + h3[i] * d * d + b2[c];
}

// ---------------------------------------------------------------------------

extern "C" void kernel_launch(void* const* d_in, const int* in_sizes, int n_in,
                              void* d_out, int out_size, void* d_ws, size_t ws_size,
                              hipStream_t stream) {
  const float* x       = (const float*)d_in[0];
  const long long* ei  = (const long long*)d_in[1];   // int64 per reference
  const float* W1      = (const float*)d_in[2];
  const float* b1      = (const float*)d_in[3];
  const float* W2      = (const float*)d_in[4];
  const float* b2      = (const float*)d_in[5];

  const int       N = in_sizes[0] / IN_C;
  const long long E = (long long)in_sizes[1] / 2;
  const long long* src = ei;
  const long long* dst = ei + E;

  // workspace (fp32): dinv[N] | h1[N*32] | agg1[N*32] | h3[N*16] | Wp1 | Wp2
  float* dinv = (float*)d_ws;
  float* h1   = dinv + N;
  float* agg1 = h1 + (size_t)N * HID_C;
  float* h3   = agg1 + (size_t)N * HID_C;
  float* Wp1  = h3 + (size_t)N * OUT_C;               // IN_C*HID_C  = 4096 f
  float* Wp2  = Wp1 + IN_C * HID_C;                   // HID_C*OUT_C = 512 f
  float* out  = (float*)d_out;

  hipMemsetAsync(agg1, 0, (size_t)N * HID_C * sizeof(float), stream);
  hipMemsetAsync(out,  0, (size_t)N * OUT_C * sizeof(float), stream);

  const int TB = 256;
  const int nodeBlocks = (N + TB - 1) / TB;

  // one-time weight repack into WMMA fragment order
  pack_wmma_b<IN_C, 2><<<(IN_C / 4 * 2 * 32 + TB - 1) / TB, TB, 0, stream>>>(W1, Wp1);
  pack_wmma_b<HID_C, 1><<<(HID_C / 4 * 1 * 32 + TB - 1) / TB, TB, 0, stream>>>(W2, Wp2);

  // symmetric normalization (deg computed with self-loops)
  fill_ones<<<nodeBlocks, TB, 0, stream>>>(dinv, N);
  deg_accum<<<(int)((E + TB - 1) / TB), TB, 0, stream>>>(dst, dinv, E);
  rsqrt_inplace<<<nodeBlocks, TB, 0, stream>>>(dinv, N);

  // layer 1: GEMM (WMMA) -> edge scatter -> self-loop + bias + relu
  const int rowTiles = (N + 15) / 16;                 // 6250 for N=100000
  const int gemmBlocks = (rowTiles + 7) / 8;          // 8 waves / block
  gcn_gemm_wmma<IN_C, 2><<<gemmBlocks, TB, 0, stream>>>(x, Wp1, h1, N);
  edge_aggregate<8><<<(int)((E * 8 + TB - 1) / TB), TB, 0, stream>>>(
      h1, src, dst, dinv, agg1, E);
  epilogue_l1<<<(int)(((long long)N * HID_C + TB - 1) / TB), TB, 0, stream>>>(
      agg1, h1, dinv, b1, N);

  // layer 2: GEMM (WMMA) -> edge scatter into d_out -> self-loop + bias
  gcn_gemm_wmma<HID_C, 1><<<gemmBlocks, TB, 0, stream>>>(agg1, Wp2, h3, N);
  edge_aggregate<4><<<(int)((E * 4 + TB - 1) / TB), TB, 0, stream>>>(
      h3, src, dst, dinv, out, E);
  epilogue_l2<<<(int)(((long long)N * OUT_C + TB - 1) / TB), TB, 0, stream>>>(
      out, h3, dinv, b2, N);
}